// CapsuleLayer_20126216749271
// MI455X (gfx1250) — compile-verified
//
#include <hip/hip_runtime.h>

// CDNA5 / gfx1250 capsule dynamic routing, fully fused (no u_hat materialization).
// u_hat recomputed per iteration via V_WMMA_F32_16X16X4_F32 (M=b tile, N=o, K=i).

typedef __attribute__((ext_vector_type(2))) float v2f;
typedef __attribute__((ext_vector_type(8))) float v8f;

#define B_ 128
#define C_ 10
#define R_ 8192
#define I_ 8
#define O_ 16

__device__ __constant__ float kLog2e = 1.4426950408889634f;

__global__ void caps_zero_f32(float* __restrict__ p, int n) {
    int i = blockIdx.x * blockDim.x + threadIdx.x;
    if (i < n) p[i] = 0.0f;
}

// One wave handles one (b-tile of 16, r-chunk of 16).
// blockDim = 256 (8 waves). gridDim = (64 r-groups, 8 b-tiles).
__launch_bounds__(256)
__global__ void caps_route_accum(const float* __restrict__ x,      // [B,R,I]
                                 const float* __restrict__ W,      // [C,R,I,O]
                                 const float* __restrict__ vsum,   // [B,C,O]
                                 float* __restrict__ S) {          // [B,C,O] accum
    __shared__ float vs[16 * C_ * O_];   // vsum tile for this b-tile, pre-scaled by log2(e)

    const int btile = blockIdx.y;
    const int b0    = btile * 16;

    for (int t = threadIdx.x; t < 16 * C_ * O_; t += blockDim.x) {
        vs[t] = vsum[(size_t)b0 * C_ * O_ + t] * kLog2e;
    }
    __syncthreads();

    const int wave = threadIdx.x >> 5;
    const int lane = threadIdx.x & 31;
    const int hf   = lane >> 4;      // which 16-lane half
    const int o    = lane & 15;      // N index (output capsule dim)
    const int m    = lane & 15;      // A-fragment M row (local b)

    const int rbase = (blockIdx.x * 8 + wave) * 16;

    // Per-lane element offsets inside a (c,r) W block of 128 contiguous floats,
    // matching the B-fragment layout of V_WMMA_F32_16X16X4_F32:
    //   vgpr0: K = (hf?2:0), vgpr1: K = (hf?3:1); second WMMA covers i=4..7.
    const int i00 = (hf ? 2 : 0), i01 = (hf ? 3 : 1);
    const int i10 = (hf ? 6 : 4), i11 = (hf ? 7 : 5);
    const int wo0 = i00 * O_ + o;
    const int wo1 = i01 * O_ + o;
    const int wo2 = i10 * O_ + o;
    const int wo3 = i11 * O_ + o;

    v8f acc[C_] = {};   // Sum over r of softmax-weighted u_hat tiles

    for (int rr = 0; rr < 16; ++rr) {
        const int r = rbase + rr;

        // A fragments: x[b0+m, r, i] (A layout: lanes=M, halves/vgprs=K)
        const float* xb = x + ((size_t)(b0 + m) * R_ + r) * I_;
        v2f a0, a1;
        a0.x = xb[i00]; a0.y = xb[i01];
        a1.x = xb[i10]; a1.y = xb[i11];

        float Z[8] = {0, 0, 0, 0, 0, 0, 0, 0};
        v8f   T[C_];

#pragma unroll
        for (int c = 0; c < C_; ++c) {
            const float* wb = W + ((size_t)c * R_ + r) * (I_ * O_);
            v2f bf0, bf1;
            bf0.x = wb[wo0]; bf0.y = wb[wo1];
            bf1.x = wb[wo2]; bf1.y = wb[wo3];

            v8f d = {};
            d = __builtin_amdgcn_wmma_f32_16x16x4_f32(false, a0, false, bf0,
                                                      (short)0, d, false, false);
            d = __builtin_amdgcn_wmma_f32_16x16x4_f32(false, a1, false, bf1,
                                                      (short)0, d, false, false);
            // d[g] = u_hat[b0 + g + 8*hf, c, r, o]

#pragma unroll
            for (int g = 0; g < 8; ++g) {
                // logit (in log2 units): sum over o of u_hat * vsum*log2e
                float p = d[g] * vs[((g + 8 * hf) * C_ + c) * O_ + o];
                p += __shfl_xor(p, 1);
                p += __shfl_xor(p, 2);
                p += __shfl_xor(p, 4);
                p += __shfl_xor(p, 8);   // broadcast within each half
                float e = exp2f(p);      // == exp(logit); logits are small, no max-sub
                Z[g] += e;
                T[c][g] = d[g] * e;
            }
        }

        // Normalize this r's contribution by its softmax denominator, accumulate.
#pragma unroll
        for (int g = 0; g < 8; ++g) {
            float inv = __builtin_amdgcn_rcpf(Z[g]);
#pragma unroll
            for (int c = 0; c < C_; ++c) acc[c][g] += T[c][g] * inv;
        }
    }

    // Flush partial s_j to global with float atomics.
#pragma unroll
    for (int c = 0; c < C_; ++c) {
#pragma unroll
        for (int g = 0; g < 8; ++g) {
            const int b = b0 + g + 8 * hf;
            atomicAdd(&S[((size_t)b * C_ + c) * O_ + o], acc[c][g]);
        }
    }
}

// One thread per (b,c): v = squash(s); out = v; vsum += v.
__global__ void caps_squash(const float* __restrict__ S,
                            float* __restrict__ vsum,
                            float* __restrict__ out) {
    int idx = blockIdx.x * blockDim.x + threadIdx.x;
    if (idx >= B_ * C_) return;
    const float* s = S + (size_t)idx * O_;
    float v[O_], sq = 0.0f;
#pragma unroll
    for (int k = 0; k < O_; ++k) { v[k] = s[k]; sq += v[k] * v[k]; }
    const float scale = (sq / (1.0f + sq)) * rsqrtf(sq + 1e-8f);
    float* vo = vsum + (size_t)idx * O_;
    float* op = out + (size_t)idx * O_;
#pragma unroll
    for (int k = 0; k < O_; ++k) {
        float vv = scale * v[k];
        op[k] = vv;
        vo[k] += vv;
    }
}

extern "C" void kernel_launch(void* const* d_in, const int* in_sizes, int n_in,
                              void* d_out, int out_size, void* d_ws, size_t ws_size,
                              hipStream_t stream) {
    const float* x = (const float*)d_in[0];        // [128, 8192, 8]
    const float* W = (const float*)d_in[1];        // [10, 8192, 8, 16]
    float* out = (float*)d_out;                    // [128, 10, 16]

    const int SN = B_ * C_ * O_;                   // 20480
    float* S    = (float*)d_ws;                    // s_j accumulator
    float* vsum = S + SN;                          // sum of v over past iterations

    (void)in_sizes; (void)n_in; (void)out_size; (void)ws_size;

    caps_zero_f32<<<(SN + 255) / 256, 256, 0, stream>>>(vsum, SN);

    for (int it = 0; it < 3; ++it) {
        caps_zero_f32<<<(SN + 255) / 256, 256, 0, stream>>>(S, SN);
        dim3 grid(64, 8);   // 512 r-chunks of 16, 8 b-tiles of 16
        caps_route_accum<<<grid, 256, 0, stream>>>(x, W, vsum, S);
        caps_squash<<<(B_ * C_ + 255) / 256, 256, 0, stream>>>(S, vsum, out);
    }
}